// DoubleFeatureTransformerSlice_46660524703858
// MI455X (gfx1250) — compile-verified
//
#include <hip/hip_runtime.h>

// -----------------------------------------------------------------------------
// NNUE double feature-transformer slice for MI455X (gfx1250).
//
// Bandwidth-bound sparse row-gather + weighted sum (~2.1 GB gathered; weight
// matrix is 80 MB and L2-resident). WMMA cannot express it (each batch row has
// a distinct gathered B matrix), so we use the CDNA5 Tensor Data Mover in
// GATHER MODE: per (slice, batch-row) workgroup, wave 0 issues two
// TENSOR_LOAD_TO_LDS gathers (16 rows x 2 KB, 16-bit row indices). TDM ops
// from one wave complete in order, so we drain in two phases
// (s_wait_tensorcnt 1 -> compute k=0..15 while rows 16..31 stream in ->
// s_wait_tensorcnt 0 -> compute k=16..31), overlapping DMA with compute.
//
// NOTE: the TDM builtin receives the LDS address only as an integer inside the
// descriptor, so LLVM cannot see that it writes the __shared__ array. We (a)
// escape the array via ptrtoint into the descriptor and (b) launder the read
// pointer through an asm memory-clobber before the DS reads, otherwise the
// compiler proves the reads undef and deletes the whole FMA loop (observed in
// round 2: ds=0, valu=5).
// -----------------------------------------------------------------------------

typedef unsigned int u32x4 __attribute__((ext_vector_type(4)));
typedef int          i32x8 __attribute__((ext_vector_type(8)));
typedef int          i32x4 __attribute__((ext_vector_type(4)));
typedef float        f32x4 __attribute__((ext_vector_type(4)));

typedef __attribute__((address_space(3))) const f32x4 lds_f32x4;

#define NUM_OUT 512   // output features per slice (row length of weight)
#define KFEAT   32    // max active features per batch row
#define THREADS 128   // 4 waves; each thread owns 4 output columns

// Pack 8 consecutive 32-bit row indices into four 16-bit pairs (D# groups 2/3
// gather-mode layout: row_index_0 in bits [15:0] of word 0, etc.).
__device__ __forceinline__ i32x4 pack_idx8(const int* __restrict__ idx) {
  i32x4 r;
  r[0] = (int)(((unsigned)idx[0] & 0xFFFFu) | ((unsigned)idx[1] << 16));
  r[1] = (int)(((unsigned)idx[2] & 0xFFFFu) | ((unsigned)idx[3] << 16));
  r[2] = (int)(((unsigned)idx[4] & 0xFFFFu) | ((unsigned)idx[5] << 16));
  r[3] = (int)(((unsigned)idx[6] & 0xFFFFu) | ((unsigned)idx[7] << 16));
  return r;
}

// Issue one TDM gather of 16 rows (each NUM_OUT f32) of W into LDS at byte
// offset lds_addr. Row indices packed 16-bit in g2 (rows 0-7) / g3 (8-15).
__device__ __forceinline__ void tdm_gather16(const float* __restrict__ W,
                                             unsigned lds_addr,
                                             unsigned num_inputs,
                                             i32x4 g2, i32x4 g3) {
  const unsigned long long base = (unsigned long long)(uintptr_t)W;

  // D# group 0: count=1 (valid), gather_index_size=0 (16-bit), gather_mode=1,
  // lds_addr, global_addr[56:0], type=2 ("image").
  u32x4 g0;
  g0[0] = 1u | (1u << 31);
  g0[1] = lds_addr;
  g0[2] = (unsigned)base;
  g0[3] = (unsigned)((base >> 32) & 0x1FFFFFFu) | (2u << 30);

  // D# group 1: workgroup_mask=0 (not in cluster), data_size=2 (4 bytes),
  // no barrier/iterate/pad. tensor_dim0 = NUM_OUT (row width, elements),
  // tensor_dim1 = num_inputs (row count), tile_dim0 = NUM_OUT (full row),
  // tile_dim1 = 16 (#valid gather indices), tensor_dim0_stride = NUM_OUT.
  i32x8 g1;
  g1[0] = (int)(2u << 16);                  // data_size = 4 bytes
  g1[1] = (int)((unsigned)NUM_OUT << 16);   // tensor_dim0[15:0] in bits[31:16]
  g1[2] = (int)(num_inputs << 16);          // tensor_dim1[15:0] in bits[31:16]
  g1[3] = (int)((unsigned)NUM_OUT << 16);   // tile_dim0 in bits[31:16]
  g1[4] = 16;                               // tile_dim1 = #valid indices
  g1[5] = NUM_OUT;                          // tensor_dim0_stride[31:0]
  g1[6] = 0;
  g1[7] = 0;

  // This toolchain's 6-arg form: (g0, g1, g2, g3, g4, cpol); group 4 unused
  // for a <=2D gather descriptor -> zero-filled.
  i32x8 g4 = {0, 0, 0, 0, 0, 0, 0, 0};

  __builtin_amdgcn_tensor_load_to_lds(g0, g1, g2, g3, g4, /*cpol=*/0);
}

__global__ __launch_bounds__(THREADS) void nnue_ft_tdm_kernel(
    const int* __restrict__ fi0, const float* __restrict__ fv0,
    const int* __restrict__ fi1, const float* __restrict__ fv1,
    const float* __restrict__ W, const float* __restrict__ bias,
    float* __restrict__ out, int batch, int num_inputs) {
  __shared__ float lds[KFEAT * NUM_OUT];  // 32 rows x 2 KB = 64 KB

  // Low 32 bits of a generic pointer into LDS are the LDS byte offset
  // (ISA flat-to-LDS aperture mapping). This also escapes the
  // array so alias analysis must assume the TDM ops may write it.
  unsigned lds_base = (unsigned)(uintptr_t)(void*)&lds[0];

  const int row   = blockIdx.x;
  const int slice = blockIdx.y;
  const int*   __restrict__ fi = (slice ? fi1 : fi0) + (size_t)row * KFEAT;
  const float* __restrict__ fv = (slice ? fv1 : fv0) + (size_t)row * KFEAT;

  const unsigned tid = threadIdx.x;
  // readfirstlane => SGPR => guaranteed scalar branch: exactly one wave issues
  // the TDM ops (TDM ignores EXEC, so EXEC-masking alone would duplicate DMAs).
  const unsigned wave = ((unsigned)__builtin_amdgcn_readfirstlane((int)tid)) >> 5;

  if (wave == 0) {
    int idx[KFEAT];
#pragma unroll
    for (int k = 0; k < KFEAT; ++k) idx[k] = fi[k];  // uniform -> s_load_b512

    const i32x4 a2 = pack_idx8(idx + 0),  a3 = pack_idx8(idx + 8);
    const i32x4 b2 = pack_idx8(idx + 16), b3 = pack_idx8(idx + 24);

    tdm_gather16(W, lds_base,                      (unsigned)num_inputs, a2, a3);
    tdm_gather16(W, lds_base + 16u * NUM_OUT * 4u, (unsigned)num_inputs, b2, b3);
    __builtin_amdgcn_s_wait_tensorcnt(1);  // in-order: first 16 rows landed
  }
  __syncthreads();  // publish rows 0..15 to all 4 waves

  // Launder the LDS base through an opaque asm with a memory clobber so the
  // staged rows count as written-by-someone; reads stay real ds_load_b128.
  asm volatile("" : "+s"(lds_base) : : "memory");
  const lds_f32x4* __restrict__ lrow =
      (const lds_f32x4*)(uintptr_t)(lds_base + (unsigned)tid * 16u);

  // Phase 1: k = 0..15 while the TDM engine streams rows 16..31.
  f32x4 acc = *(const f32x4*)(bias + tid * 4);
#pragma unroll
  for (int k = 0; k < KFEAT / 2; ++k) {
    const float v = fv[k];                              // uniform -> SGPR
    const f32x4 w = lrow[(unsigned)k * (NUM_OUT / 4)];  // ds_load_b128
    acc.x = fmaf(v, w.x, acc.x);
    acc.y = fmaf(v, w.y, acc.y);
    acc.z = fmaf(v, w.z, acc.z);
    acc.w = fmaf(v, w.w, acc.w);
  }

  if (wave == 0) __builtin_amdgcn_s_wait_tensorcnt(0);  // rows 16..31 landed
  __syncthreads();  // publish rows 16..31 to all 4 waves

  // Phase 2: k = 16..31.
#pragma unroll
  for (int k = KFEAT / 2; k < KFEAT; ++k) {
    const float v = fv[k];
    const f32x4 w = lrow[(unsigned)k * (NUM_OUT / 4)];
    acc.x = fmaf(v, w.x, acc.x);
    acc.y = fmaf(v, w.y, acc.y);
    acc.z = fmaf(v, w.z, acc.z);
    acc.w = fmaf(v, w.w, acc.w);
  }

  // Output is write-once streaming data: non-temporal store keeps the 64 MB
  // result from evicting the L2-resident 80 MB weight matrix.
  float* o = out + ((size_t)slice * (size_t)batch + (size_t)row) * NUM_OUT
                 + (size_t)tid * 4;
  __builtin_nontemporal_store(acc, (f32x4*)o);  // global_store_b128 (NT)
}

extern "C" void kernel_launch(void* const* d_in, const int* in_sizes, int n_in,
                              void* d_out, int out_size, void* d_ws, size_t ws_size,
                              hipStream_t stream) {
  const int*   fi0  = (const int*)d_in[0];
  const float* fv0  = (const float*)d_in[1];
  const int*   fi1  = (const int*)d_in[2];
  const float* fv1  = (const float*)d_in[3];
  const float* W    = (const float*)d_in[4];
  const float* bias = (const float*)d_in[5];
  float* out = (float*)d_out;

  const int batch      = in_sizes[0] / KFEAT;    // 16384
  const int num_inputs = in_sizes[4] / NUM_OUT;  // 40960

  dim3 grid((unsigned)batch, 2u, 1u);
  hipLaunchKernelGGL(nnue_ft_tdm_kernel, grid, dim3(THREADS, 1, 1), 0, stream,
                     fi0, fv0, fi1, fv1, W, bias, out, batch, num_inputs);
}